// SocialEncoder_62345745268917
// MI455X (gfx1250) — compile-verified
//
#include <hip/hip_runtime.h>

// ---------------------------------------------------------------------------
// SocialEncoder on MI455X (gfx1250):
//   emb = relu(x @ W1^T + b1)          (parallel, producer waves, f32 WMMA)
//   h_T = LSTM(emb)                     (serial scan, 2 consumer waves, f32 WMMA)
//   out[i,j,:] = h_T[i,:]               (the reference "sort" is an identity
//                                        gather over a broadcast -> distances
//                                        and train_len are dead inputs)
// Single workgroup, 8 waves (wave32). Producer/consumer double-buffered LDS
// pipeline, one barrier per 8-timestep chunk.
// ---------------------------------------------------------------------------

#define T_LEN 16384
#define NAG   32
#define HID   16
#define CH    8                      // timesteps per pipeline chunk
#define NC    (T_LEN / CH)           // 2048 chunks
#define XFRAG (CH * 2 * 4 * 8 * 32)  // floats per ring buffer = 16384 (64 KB)

typedef float v2f __attribute__((ext_vector_type(2)));
typedef float v8f __attribute__((ext_vector_type(8)));

// D = A(16x4 f32) * B(4x16 f32) + C(16x16 f32), full fp32 WMMA
static __device__ inline v8f wmma4(v2f a, v2f b, v8f c) {
  return __builtin_amdgcn_wmma_f32_16x16x4_f32(false, a, false, b,
                                               (short)0, c, false, false);
}

static __device__ inline float fexp(float x) {
  return __builtin_amdgcn_exp2f(x * 1.4426950408889634f);
}
#if __has_builtin(__builtin_amdgcn_tanhf)
static __device__ inline float tanhx(float x) { return __builtin_amdgcn_tanhf(x); }
#else
static __device__ inline float tanhx(float x) {
  // 1 - 2/(e^{2x}+1); saturates correctly at +/-1
  return fmaf(-2.0f, __builtin_amdgcn_rcpf(fexp(2.0f * x) + 1.0f), 1.0f);
}
#endif
static __device__ inline float sigm(float x) {
  return fmaf(0.5f, tanhx(0.5f * x), 0.5f);
}

__global__ __launch_bounds__(256, 1)
void social_lstm_kernel(const float* __restrict__ xseq,
                        const float* __restrict__ W1,
                        const float* __restrict__ b1,
                        const float* __restrict__ Wih,
                        const float* __restrict__ Whh,
                        const float* __restrict__ bih,
                        const float* __restrict__ bhh,
                        float* __restrict__ out) {
  // Double-buffered x-projection ring, stored in exact WMMA C-fragment layout:
  //   xbuf[par][ (((tl*2+half)*4+nt)*8 + r)*32 + lane ]
  __shared__ float xbuf[2 * XFRAG];
  // Per-consumer-wave h tile, row stride 17 floats -> bank-conflict free
  __shared__ float hbuf[2][16 * 17];

  const int tid  = threadIdx.x;
  const int wave = tid >> 5;
  const int lane = tid & 31;
  const int l15  = lane & 15;
  const int kHi  = (lane & 16) ? 2 : 0;   // A/B fragment K-subcolumn select

  // zero initial hidden state (h0 = 0); visible to consumers at first barrier
  for (int i = tid; i < 2 * 16 * 17; i += 256) ((float*)hbuf)[i] = 0.0f;

  if (wave < 2) {
    // ================= CONSUMER: LSTM recurrence, agents [w*16, w*16+16) ====
    const int w = wave;
    // B fragments of W_hh^T : B[k][n] = Whh[n][k]
    v2f Bh[4][4];
#pragma unroll
    for (int nt = 0; nt < 4; ++nt) {
      const int n = nt * 16 + l15;
#pragma unroll
      for (int kb = 0; kb < 4; ++kb) {
        const int k0 = kb * 4 + kHi;
        v2f t; t.x = Whh[n * 16 + k0]; t.y = Whh[n * 16 + k0 + 1];
        Bh[nt][kb] = t;
      }
    }
    float c8[8];                         // cell state, resident in D-layout
#pragma unroll
    for (int r = 0; r < 8; ++r) c8[r] = 0.0f;
    const int rowBase = (lane >= 16) ? 8 : 0;
    float* hb = &hbuf[w][0];

    for (int chunk = 0; chunk < NC; ++chunk) {
      __syncthreads();                   // xbuf[par] for this chunk is ready
      const int par = chunk & 1;
      for (int tl = 0; tl < CH; ++tl) {
        // A fragments from h (transpose via LDS; same-wave DS ops in-order)
        v2f A[4];
#pragma unroll
        for (int kb = 0; kb < 4; ++kb) {
          const int k0 = kb * 4 + kHi;
          v2f t; t.x = hb[l15 * 17 + k0]; t.y = hb[l15 * 17 + k0 + 1];
          A[kb] = t;
        }
        v8f acc[4];                      // i, f, g, o gate pre-activations
#pragma unroll
        for (int nt = 0; nt < 4; ++nt) {
          const int base = par * XFRAG + (((tl * 2 + w) * 4 + nt) * 8) * 32 + lane;
          v8f a;
#pragma unroll
          for (int r = 0; r < 8; ++r) a[r] = xbuf[base + r * 32];  // C := xproj
#pragma unroll
          for (int kb = 0; kb < 4; ++kb) a = wmma4(A[kb], Bh[nt][kb], a);
          acc[nt] = a;
        }
#pragma unroll
        for (int r = 0; r < 8; ++r) {
          const float ig = sigm(acc[0][r]);
          const float fg = sigm(acc[1][r]);
          const float gg = tanhx(acc[2][r]);
          const float og = sigm(acc[3][r]);
          const float cn = fmaf(fg, c8[r], ig * gg);
          c8[r] = cn;
          hb[(rowBase + r) * 17 + l15] = og * tanhx(cn);
        }
      }
    }
  } else {
    // ================= PRODUCER: fused MLP + x-projection ===================
    // per-lane W1/b1 slices matching A-fragment k positions
    float pw0[8], pw1[8], pb[8];
#pragma unroll
    for (int j = 0; j < 8; ++j) {
      const int k = (j >> 1) * 4 + kHi + (j & 1);
      pw0[j] = W1[2 * k]; pw1[j] = W1[2 * k + 1]; pb[j] = b1[k];
    }
    // B fragments of W_ih^T, and fused bias (b_ih + b_hh)
    v2f Bi[4][4]; float biasn[4];
#pragma unroll
    for (int nt = 0; nt < 4; ++nt) {
      const int n = nt * 16 + l15;
      biasn[nt] = bih[n] + bhh[n];
#pragma unroll
      for (int kb = 0; kb < 4; ++kb) {
        const int k0 = kb * 4 + kHi;
        v2f t; t.x = Wih[n * 16 + k0]; t.y = Wih[n * 16 + k0 + 1];
        Bi[nt][kb] = t;
      }
    }

    // produce one chunk into parity pp
    auto produce = [&](int cc, int pp) {
      for (int tile = wave - 2; tile < CH * 2; tile += 6) {
        const int tl = tile >> 1, hf = tile & 1;
        const int agent = hf * 16 + l15;
        const int t = cc * CH + tl;
        const float* xp = xseq + ((size_t)t * NAG + agent) * 2;
        const float x0 = xp[0], x1 = xp[1];
        v2f A[4];                        // emb rows, built directly in A layout
#pragma unroll
        for (int kb = 0; kb < 4; ++kb) {
          const float e0 =
              fmaxf(fmaf(x1, pw1[2 * kb], fmaf(x0, pw0[2 * kb], pb[2 * kb])), 0.0f);
          const float e1 =
              fmaxf(fmaf(x1, pw1[2 * kb + 1], fmaf(x0, pw0[2 * kb + 1], pb[2 * kb + 1])), 0.0f);
          v2f t2; t2.x = e0; t2.y = e1; A[kb] = t2;
        }
#pragma unroll
        for (int nt = 0; nt < 4; ++nt) {
          v8f a;
#pragma unroll
          for (int r = 0; r < 8; ++r) a[r] = biasn[nt];   // C := bias
#pragma unroll
          for (int kb = 0; kb < 4; ++kb) a = wmma4(A[kb], Bi[nt][kb], a);
          const int base = pp * XFRAG + (((tl * 2 + hf) * 4 + nt) * 8) * 32 + lane;
#pragma unroll
          for (int r = 0; r < 8; ++r) xbuf[base + r * 32] = a[r];
        }
      }
    };

    produce(0, 0);                       // prologue: fill ring buffer 0
    for (int chunk = 0; chunk < NC; ++chunk) {
      __syncthreads();                   // consumers may now read par=chunk&1
      if (chunk + 1 < NC) produce(chunk + 1, (chunk + 1) & 1);
    }
  }

  __syncthreads();
  // out[i,j,k] = h_final[i,k]  (the reference's sort-gather is an identity)
  for (int idx = tid; idx < NAG * NAG * HID; idx += 256) {
    const int i = idx >> 9;        // agent
    const int k = idx & 15;        // hidden unit
    out[idx] = hbuf[i >> 4][(i & 15) * 17 + k];
  }
}

extern "C" void kernel_launch(void* const* d_in, const int* in_sizes, int n_in,
                              void* d_out, int out_size, void* d_ws, size_t ws_size,
                              hipStream_t stream) {
  (void)in_sizes; (void)n_in; (void)out_size; (void)d_ws; (void)ws_size;
  const float* xseq = (const float*)d_in[0];
  // d_in[1] (distances) and d_in[2] (train_len) are dead: the reference's
  // take_along_axis gathers from a tensor broadcast along the gathered axis.
  const float* W1  = (const float*)d_in[3];
  const float* b1  = (const float*)d_in[4];
  const float* Wih = (const float*)d_in[5];
  const float* Whh = (const float*)d_in[6];
  const float* bih = (const float*)d_in[7];
  const float* bhh = (const float*)d_in[8];
  float* out = (float*)d_out;
  hipLaunchKernelGGL(social_lstm_kernel, dim3(1), dim3(256), 0, stream,
                     xseq, W1, b1, Wih, Whh, bih, bhh, out);
}